// ScalingLayer_33835752358522
// MI455X (gfx1250) — compile-verified
//
#include <hip/hip_runtime.h>

// Native clang vector type so __builtin_nontemporal_load/store accept it.
typedef float f32x4 __attribute__((ext_vector_type(4)));
// GCC-style int vector matching the async builtin's parameter type exactly:
//   'int __attribute__((__vector_size__(4 * sizeof(int)))) *'
typedef int v4i __attribute__((vector_size(16)));

#define N_COLS        4096
#define N_ROWS        8192
#define NC4           (N_COLS / 4)     // 1024 float4 columns
#define TPB           256              // 8 wave32 per block
#define ROWS_PER_TILE 32
#define RCHUNK        4                // rows per async double-buffer group
#define NGROUPS       (ROWS_PER_TILE / RCHUNK)
#define PF_ROWS       8                // prefetch distance (fallback path)

// ---- CDNA5 async global->LDS path, hedged on builtin availability ---------
#if defined(__HIP_DEVICE_COMPILE__) && \
    __has_builtin(__builtin_amdgcn_global_load_async_to_lds_b128)
#define USE_ASYNC 1
#else
#define USE_ASYNC 0
#endif

#if USE_ASYNC
// (global src, lds dst, imm offset, imm cpol). Parameters are generic v4i*
// per the compiler diagnostic; codegen handles the p1/p3 address spaces.
#define ASYNC_COPY_B128(gsrc, ldst)                                     \
    __builtin_amdgcn_global_load_async_to_lds_b128(                     \
        (v4i*)(gsrc), (v4i*)(ldst), 0, 0)

#if __has_builtin(__builtin_amdgcn_s_wait_asynccnt)
#define WAIT_ASYNC(n) __builtin_amdgcn_s_wait_asynccnt(n)
#else
#define WAIT_ASYNC(n) asm volatile("s_wait_asynccnt %0" ::"i"(n))
#endif

// WAR-hazard fence: async LDS writes are unordered with DS ops (ISA §10.7),
// so before re-targeting a buffer that earlier ds_loads read, drain DScnt.
// Also acts as a compiler barrier so the async issues can't be hoisted above
// the preceding LDS reads. In steady state DScnt is already 0 here.
#define WAIT_DS0() asm volatile("s_wait_dscnt 0x0" ::: "memory")
#endif // USE_ASYNC

// ---------------------------------------------------------------------------
// Kernel 1: gather diag(W) into workspace. 4096 loads, stride 4097*4 B —
// touches 4096 distinct cachelines of the 64 MB W exactly once. Trivial cost.
// ---------------------------------------------------------------------------
__global__ __launch_bounds__(TPB)
void ScalingLayer_diag_gather(const float* __restrict__ W,
                              float* __restrict__ scale) {
    const int j = blockIdx.x * TPB + threadIdx.x;
    if (j < N_COLS) {
        scale[j] = W[(size_t)j * (size_t)(N_COLS + 1)];
    }
}

// ---------------------------------------------------------------------------
// Kernel 2: streaming out[r, c] = x[r, c] * scale[c] + b[c].
//
// Memory-bound: 268 MB total / 23.3 TB/s => ~11.5 us floor. Thread t owns 4
// fixed columns, loops over 32 rows; scale/bias are loaded once per thread
// (RT policy, 32 KB resident in L2). The 256 MB x/out stream must not thrash
// the 192 MB global L2, so output stores are non-temporal.
//
// ASYNC path: x is staged through LDS with GLOBAL_LOAD_ASYNC_TO_LDS_B128 in
// a 2-deep double buffer of 4-row groups (32 KB LDS/block; 8 blocks/WGP fits
// the 320 KB WGP LDS). Each lane's async copies target LDS addresses derived
// from its own VGPRs and retire in order on its own wave's ASYNCcnt, so
// `s_wait_asynccnt RCHUNK` after issuing group g+1 guarantees group g is
// resident — no workgroup barriers needed anywhere.
// ---------------------------------------------------------------------------
__global__ __launch_bounds__(TPB)
void ScalingLayer_scale_bias_stream(const float* __restrict__ x,
                                    const float* __restrict__ scale,
                                    const float* __restrict__ bias,
                                    float* __restrict__ out) {
    const int tid = threadIdx.x;
    const int c4  = blockIdx.x * TPB + tid;          // [0, NC4)

    f32x4* __restrict__ x4 = (f32x4*)x;              // const dropped: read-only use
    f32x4* __restrict__ o4 = (f32x4*)out;

    // One-time per-thread loads; stay cached (RT policy) across the dispatch.
    const f32x4 s  = ((const f32x4*)scale)[c4];
    const f32x4 bb = ((const f32x4*)bias)[c4];

    const size_t base =
        (size_t)blockIdx.y * (size_t)ROWS_PER_TILE * NC4 + (size_t)c4;

#if USE_ASYNC
    __shared__ f32x4 buf[2][RCHUNK][TPB];            // 2 * 16 KB

    // Prologue: fill buffer 0 (4 async b128 copies per lane).
#pragma unroll
    for (int rr = 0; rr < RCHUNK; ++rr) {
        ASYNC_COPY_B128(&x4[base + (size_t)rr * NC4], &buf[0][rr][tid]);
    }

    for (int g = 0; g < NGROUPS; ++g) {
        if (g + 1 < NGROUPS) {
            // Ensure the LDS reads that previously consumed this buffer have
            // fully retired before the DMA may overwrite it (WAR fence).
            WAIT_DS0();
            const size_t nbase = base + (size_t)(g + 1) * RCHUNK * NC4;
#pragma unroll
            for (int rr = 0; rr < RCHUNK; ++rr) {
                ASYNC_COPY_B128(&x4[nbase + (size_t)rr * NC4],
                                &buf[(g + 1) & 1][rr][tid]);
            }
            WAIT_ASYNC(RCHUNK);   // 8 in flight -> first 4 (group g) done
        } else {
            WAIT_ASYNC(0);        // drain for the final group
        }

        const size_t obase = base + (size_t)g * RCHUNK * NC4;
#pragma unroll
        for (int rr = 0; rr < RCHUNK; ++rr) {
            f32x4 v = buf[g & 1][rr][tid];           // ds_load_b128
            f32x4 o;
            o.x = fmaf(v.x, s.x, bb.x);
            o.y = fmaf(v.y, s.y, bb.y);
            o.z = fmaf(v.z, s.z, bb.z);
            o.w = fmaf(v.w, s.w, bb.w);
            __builtin_nontemporal_store(o, &o4[obase + (size_t)rr * NC4]);
        }
    }
#else
    // Fallback: direct non-temporal b128 stream with software prefetch.
    size_t idx = base;
#pragma unroll 8
    for (int r = 0; r < ROWS_PER_TILE; ++r, idx += NC4) {
        __builtin_prefetch(&x4[idx + (size_t)PF_ROWS * NC4], 0, 0);
        f32x4 v = __builtin_nontemporal_load(&x4[idx]);
        f32x4 o;
        o.x = fmaf(v.x, s.x, bb.x);
        o.y = fmaf(v.y, s.y, bb.y);
        o.z = fmaf(v.z, s.z, bb.z);
        o.w = fmaf(v.w, s.w, bb.w);
        __builtin_nontemporal_store(o, &o4[idx]);
    }
#endif
}

// ---------------------------------------------------------------------------
// Harness entry point.
//   d_in[0] = x  [8192*4096] f32
//   d_in[1] = W  [4096*4096] f32 (only diagonal used)
//   d_in[2] = b  [4096]      f32
//   d_out   = out[8192*4096] f32
//   d_ws    = scratch (4096 floats = 16 KB for diag(W))
// ---------------------------------------------------------------------------
extern "C" void kernel_launch(void* const* d_in, const int* in_sizes, int n_in,
                              void* d_out, int out_size, void* d_ws, size_t ws_size,
                              hipStream_t stream) {
    const float* x = (const float*)d_in[0];
    const float* W = (const float*)d_in[1];
    const float* b = (const float*)d_in[2];
    float* out     = (float*)d_out;
    float* scale   = (float*)d_ws;

    ScalingLayer_diag_gather<<<dim3(N_COLS / TPB), dim3(TPB), 0, stream>>>(W, scale);

    // 4 column-blocks x 256 row tiles = 1024 blocks = 8192 wave32 in flight.
    dim3 grid(NC4 / TPB, N_ROWS / ROWS_PER_TILE);
    ScalingLayer_scale_bias_stream<<<grid, dim3(TPB), 0, stream>>>(x, scale, b, out);
}